// GAT_40913858461763
// MI455X (gfx1250) — compile-verified
//
#include <hip/hip_runtime.h>
#include <hip/hip_bf16.h>

// ---------------- problem constants (match reference) ----------------
#define GN  30000      // nodes
#define GE  480000     // edges
#define GIN 256        // input feature dim
#define GD  32         // hidden per head
#define GH  4          // heads (layers 0,1)
#define GC  47         // classes (final layer, 1 head)
#define NEG_SLOPE 0.2f

typedef float v2f __attribute__((ext_vector_type(2)));
typedef float v8f __attribute__((ext_vector_type(8)));

#define CDIV(a,b) (((a)+(b)-1)/(b))

// -------- monotonic float<->uint encoding for atomic max on floats --------
__device__ __forceinline__ unsigned encf(float f) {
  unsigned u = __float_as_uint(f);
  return (u & 0x80000000u) ? ~u : (u | 0x80000000u);
}
__device__ __forceinline__ float decf(unsigned u) {
  return (u & 0x80000000u) ? __uint_as_float(u ^ 0x80000000u)
                           : __uint_as_float(~u);
}
// encf(-inf) == 0x007FFFFF
#define ENC_NEG_INF 0x007FFFFFu

// ---------------- utility fills ----------------
__global__ void fill_f32(float* p, float v, int n) {
  int i = blockIdx.x * blockDim.x + threadIdx.x;
  if (i < n) p[i] = v;
}
__global__ void fill_u32(unsigned* p, unsigned v, int n) {
  int i = blockIdx.x * blockDim.x + threadIdx.x;
  if (i < n) p[i] = v;
}
__global__ void copy_f32(const float* __restrict__ in, float* __restrict__ o, int n) {
  int i = blockIdx.x * blockDim.x + threadIdx.x;
  if (i < n) o[i] = in[i];
}

// ---------------- B transpose+pad: Bt[Npad][K] = B[K][Nn]^T (zero pad) -------
__global__ void transpose_pad_kernel(const float* __restrict__ B,
                                     float* __restrict__ Bt,
                                     int K, int Nn, int Npad) {
  int i = blockIdx.x * blockDim.x + threadIdx.x;   // over Npad*K
  if (i >= Npad * K) return;
  int j = i / K;      // column of B == row of Bt
  int k = i % K;
  Bt[i] = (j < Nn) ? B[(size_t)k * Nn + j] : 0.0f;
}

// ---------------- f32 WMMA GEMM: C[M,Nn] (+)= A[M,K] x B[K,Nn] ----------------
// B supplied pre-transposed+padded as Bt[Npad][K]. One wave per 16x16 tile
// using V_WMMA_F32_16X16X4_F32. M%16==0 (30000=1875*16), K%4==0 (256,128).
// Inner loop: two contiguous float2 loads per lane + WMMA, no guards.
__global__ void gemm_wmma_f32(const float* __restrict__ A,
                              const float* __restrict__ Bt,
                              float* __restrict__ C,
                              int M, int K, int Nn, int Npad, int accum) {
  const int lane   = threadIdx.x & 31;
  const int wave   = threadIdx.x >> 5;
  const int tile   = blockIdx.x * (blockDim.x >> 5) + wave;
  const int tilesN = Npad >> 4;
  const int tilesM = M >> 4;
  if (tile >= tilesM * tilesN) return;   // wave-uniform exit: EXEC all-1s below

  const int tm    = tile / tilesN;
  const int tn    = tile % tilesN;
  const int row   = tm * 16 + (lane & 15);        // A row for this lane
  const int col   = tn * 16 + (lane & 15);        // B/C column for this lane
  const int khalf = (lane >> 4) * 2;              // lanes 0-15 -> K 0,1 ; 16-31 -> K 2,3
  const int mbase = tm * 16 + (lane >> 4) * 8;    // C rows: vgpr i -> M = mbase+i
  const bool colok = (col < Nn);

  const float* ap = A  + (size_t)row * K + khalf; // contiguous pair per k-step
  const float* bp = Bt + (size_t)col * K + khalf; // contiguous pair per k-step

  v8f c = {};
  if (accum) {
#pragma unroll
    for (int i = 0; i < 8; ++i)
      c[i] = colok ? C[(size_t)(mbase + i) * Nn + col] : 0.0f;
  }

  for (int k = 0; k < K; k += 4) {
    v2f a = *(const v2f*)(ap + k);
    v2f b = *(const v2f*)(bp + k);
    c = __builtin_amdgcn_wmma_f32_16x16x4_f32(
        /*neg_a=*/false, a, /*neg_b=*/false, b,
        /*c_mod=*/(short)0, c, /*reuse_a=*/false, /*reuse_b=*/false);
  }

  if (colok) {
#pragma unroll
    for (int i = 0; i < 8; ++i)
      C[(size_t)(mbase + i) * Nn + col] = c[i];
  }
}

// ---------------- per-node attention projections el/er ----------------
__global__ void eler_kernel(const float* __restrict__ feat,
                            const float* __restrict__ al,
                            const float* __restrict__ ar,
                            float* __restrict__ el, float* __restrict__ er,
                            int n, int heads, int d) {
  int idx = blockIdx.x * blockDim.x + threadIdx.x;
  if (idx >= n * heads) return;
  int node = idx / heads, h = idx % heads;
  const float* f = feat + (size_t)node * heads * d + h * d;
  const float* a = al + h * d;
  const float* b = ar + h * d;
  float sl = 0.f, sr = 0.f;
  for (int i = 0; i < d; ++i) { float v = f[i]; sl += v * a[i]; sr += v * b[i]; }
  el[idx] = sl; er[idx] = sr;
}

// ---------------- edge attention score + running segment max ----------------
__global__ void edge_e_kernel(const float* __restrict__ el, const float* __restrict__ er,
                              const int* __restrict__ src, const int* __restrict__ dst,
                              float* __restrict__ eout, unsigned* __restrict__ emaxu,
                              int ne, int heads) {
  int idx = blockIdx.x * blockDim.x + threadIdx.x;
  if (idx >= ne * heads) return;
  int e = idx / heads, h = idx % heads;
  float v = el[src[e] * heads + h] + er[dst[e] * heads + h];
  v = (v >= 0.f) ? v : NEG_SLOPE * v;          // leaky_relu
  eout[idx] = v;
  atomicMax(&emaxu[dst[e] * heads + h], encf(v));
}

// ---------------- softmax denominator ----------------
__global__ void edge_wsum_kernel(const float* __restrict__ eout,
                                 const unsigned* __restrict__ emaxu,
                                 const int* __restrict__ dst,
                                 float* __restrict__ wsum, int ne, int heads) {
  int idx = blockIdx.x * blockDim.x + threadIdx.x;
  if (idx >= ne * heads) return;
  int e = idx / heads, h = idx % heads;
  int di = dst[e] * heads + h;
  float m = decf(emaxu[di]);
  atomicAdd(&wsum[di], __expf(eout[idx] - m));
}

// ---------------- weighted message scatter-add ----------------
__global__ void edge_msg_kernel(const float* __restrict__ eout,
                                const unsigned* __restrict__ emaxu,
                                const float* __restrict__ wsum,
                                const float* __restrict__ feat,
                                const int* __restrict__ src, const int* __restrict__ dst,
                                float* __restrict__ rst, int ne, int heads, int d) {
  int idx = blockIdx.x * blockDim.x + threadIdx.x;
  if (idx >= ne * heads) return;
  int e = idx / heads, h = idx % heads;
  int sn = src[e], dn = dst[e];
  int di = dn * heads + h;
  float m  = decf(emaxu[di]);
  float ws = wsum[di];
  ws = (ws > 1e-9f) ? ws : 1e-9f;
  float a = __expf(eout[idx] - m) / ws;
  const float* f = feat + (size_t)sn * heads * d + h * d;
  float*       r = rst  + (size_t)dn * heads * d + h * d;
  for (int i = 0; i < d; ++i) atomicAdd(&r[i], f[i] * a);
}

// ---------------- ELU (or ELU∘ELU) ----------------
__global__ void act_elu_kernel(const float* __restrict__ in, float* __restrict__ o,
                               int n, int twice) {
  int i = blockIdx.x * blockDim.x + threadIdx.x;
  if (i >= n) return;
  float v = in[i];
  v = (v > 0.f) ? v : expm1f(v);
  if (twice) v = (v > 0.f) ? v : expm1f(v);
  o[i] = v;
}

// ======================= host side =======================
static void launch_gemm(const float* A, const float* B, float* Cb, float* Bt,
                        int K, int Nn, int accum, hipStream_t stream) {
  const int Npad = CDIV(Nn, 16) * 16;
  transpose_pad_kernel<<<CDIV(Npad * K, 256), 256, 0, stream>>>(B, Bt, K, Nn, Npad);
  const int tiles = (GN / 16) * (Npad / 16);
  gemm_wmma_f32<<<CDIV(tiles, 8), 256, 0, stream>>>(A, Bt, Cb, GN, K, Nn, Npad, accum);
}

static void run_gat_layer(const float* hin, int Fin,
                          const float* W, const float* resW,
                          const float* al, const float* ar,
                          int heads, int d,
                          const int* src, const int* dst,
                          float* eout,                 // d_out slice [E*heads]
                          float* feat, float* rst,     // N*Fout buffers
                          float* Bt,
                          float* elb, float* erb, unsigned* emaxu, float* wsumb,
                          hipStream_t stream) {
  const int Fout = heads * d;

  // feat = hin @ W
  launch_gemm(hin, W, feat, Bt, Fin, Fout, 0, stream);

  // el/er
  eler_kernel<<<CDIV(GN * heads, 256), 256, 0, stream>>>(feat, al, ar, elb, erb,
                                                         GN, heads, d);
  // init accumulators
  fill_u32<<<CDIV(GN * heads, 256), 256, 0, stream>>>(emaxu, ENC_NEG_INF, GN * heads);
  fill_f32<<<CDIV(GN * heads, 256), 256, 0, stream>>>(wsumb, 0.f, GN * heads);
  fill_f32<<<CDIV(GN * Fout, 256), 256, 0, stream>>>(rst, 0.f, GN * Fout);

  // edge pipeline
  edge_e_kernel<<<CDIV(GE * heads, 256), 256, 0, stream>>>(elb, erb, src, dst,
                                                           eout, emaxu, GE, heads);
  edge_wsum_kernel<<<CDIV(GE * heads, 256), 256, 0, stream>>>(eout, emaxu, dst,
                                                              wsumb, GE, heads);
  edge_msg_kernel<<<CDIV(GE * heads, 256), 256, 0, stream>>>(eout, emaxu, wsumb,
                                                             feat, src, dst, rst,
                                                             GE, heads, d);
  // residual: rst += hin @ resW
  if (resW)
    launch_gemm(hin, resW, rst, Bt, Fin, Fout, 1, stream);
}

extern "C" void kernel_launch(void* const* d_in, const int* in_sizes, int n_in,
                              void* d_out, int out_size, void* d_ws, size_t ws_size,
                              hipStream_t stream) {
  const float* x     = (const float*)d_in[0];
  const float* W0    = (const float*)d_in[1];
  const float* al0   = (const float*)d_in[2];
  const float* ar0   = (const float*)d_in[3];
  const float* W1    = (const float*)d_in[4];
  const float* resW1 = (const float*)d_in[5];
  const float* al1   = (const float*)d_in[6];
  const float* ar1   = (const float*)d_in[7];
  const float* W2    = (const float*)d_in[8];
  const float* resW2 = (const float*)d_in[9];
  const float* al2   = (const float*)d_in[10];
  const float* ar2   = (const float*)d_in[11];
  const int*   src   = (const int*)d_in[12];
  const int*   dst   = (const int*)d_in[13];

  float* out    = (float*)d_out;
  float* logits = out;                         // [N, C]
  float* e0     = logits + (size_t)GN * GC;    // [E, H]
  float* e1     = e0 + (size_t)GE * GH;        // [E, H]
  float* e2     = e1 + (size_t)GE * GH;        // [E, 1]

  // workspace carving (floats): 3 big N*128 buffers + Bt + per-node scalars
  float* ws = (float*)d_ws;
  const size_t NB = (size_t)GN * (GH * GD);    // 30000*128
  float*    B0    = ws;            // feat
  float*    B1    = B0 + NB;       // rst
  float*    B2    = B1 + NB;       // h (activated node features)
  float*    Bt    = B2 + NB;       // transposed+padded weights, <= 128*256
  float*    elb   = Bt + (size_t)128 * 256;    // N*H
  float*    erb   = elb + (size_t)GN * GH;
  unsigned* emaxu = (unsigned*)(erb + (size_t)GN * GH);
  float*    wsumb = (float*)(emaxu + (size_t)GN * GH);

  // ---- layer 0: IN -> H*D, no residual; h1 = elu(rst0) ----
  run_gat_layer(x, GIN, W0, nullptr, al0, ar0, GH, GD, src, dst,
                e0, B0, B1, Bt, elb, erb, emaxu, wsumb, stream);
  act_elu_kernel<<<CDIV((int)NB, 256), 256, 0, stream>>>(B1, B2, (int)NB, 0);

  // ---- layer 1: H*D -> H*D, residual; h2 = elu(elu(rst1 + h1@resW1)) ----
  run_gat_layer(B2, GH * GD, W1, resW1, al1, ar1, GH, GD, src, dst,
                e1, B0, B1, Bt, elb, erb, emaxu, wsumb, stream);
  act_elu_kernel<<<CDIV((int)NB, 256), 256, 0, stream>>>(B1, B2, (int)NB, 1);

  // ---- layer 2: H*D -> 1*C, residual, no activation; logits = rst2 ----
  run_gat_layer(B2, GH * GD, W2, resW2, al2, ar2, 1, GC, src, dst,
                e2, B0, B1, Bt, elb, erb, emaxu, wsumb, stream);
  copy_f32<<<CDIV(GN * GC, 256), 256, 0, stream>>>(B1, logits, GN * GC);
}